// kernel_generated_2_21887153341192
// MI455X (gfx1250) — compile-verified
//
#include <hip/hip_runtime.h>

// D = A x B + C on the CDNA5 matrix pipe, full fp32 (V_WMMA_F32_16X16X4_F32).
// Problem is HBM-bound (193 MB moved / 11.1 GFLOP -> ~8.3 us at 23.3 TB/s),
// so fp32 WMMA is free accuracy; NT stores keep x resident in the 192MB L2.

typedef __attribute__((ext_vector_type(2))) float v2f;
typedef __attribute__((ext_vector_type(8))) float v8f;

#define BATCH 64
#define CIN   48
#define HH    56
#define WW    56
#define COUT  192
#define QQ    28            // H/2 : rolled axis
#define KTOT  144           // CIN * 3 taps
#define NCOL  112           // 2 * WW columns per (b,n) slab
#define NT    7             // 112/16 column tiles per wave
#define THREADS 384         // 12 waves: 12 x 16 = 192 output channels
#define SX_ELEMS (3*CIN*NCOL)   // 16128 f32 = 63 KB LDS

__global__ __launch_bounds__(THREADS)
void shiftconv_wmma_f32(const float* __restrict__ x,
                        const float* __restrict__ w,
                        float* __restrict__ out)
{
    __shared__ float sX[SX_ELEMS];   // [tap][j][o*56+m], roll+pad folded in

    const int bb = blockIdx.x / QQ;
    const int n  = blockIdx.x % QQ;

    // ---- stage 3-tap slab of x into LDS (16128 = 42*384, fully uniform) ----
    for (int idx = threadIdx.x; idx < SX_ELEMS; idx += THREADS) {
        const int tap = idx / (CIN * NCOL);
        const int r   = idx - tap * (CIN * NCOL);
        const int j   = r / NCOL;
        const int col = r - j * NCOL;
        const int o   = col / WW;
        const int m   = col - o * WW;
        const int qn  = n + tap - 1;            // index into padded 28-axis
        float v = 0.0f;
        if (qn >= 0 && qn < QQ) {               // zero-pad at n=0 / n=27
            int q = qn - 1; if (q < 0) q += QQ; // H-roll (SHIFT=1) folded in
            v = x[((size_t)(bb * CIN + j) * HH + (2 * q + o)) * WW + m];
        }
        sX[idx] = v;
    }
    __syncthreads();

    // ---- per-wave GEMM strip: 16 out-channels x 112 columns, K = 144 ----
    const int lane = threadIdx.x & 31;
    const int wave = threadIdx.x >> 5;   // 0..11
    const int half = lane >> 4;          // lane-half selects K pair (ISA 7.12.2)
    const int lm   = lane & 15;          // A: row M / B,D: column N
    const int i0   = wave * 16;

    v8f acc[NT];
    #pragma unroll
    for (int ct = 0; ct < NT; ++ct) acc[ct] = (v8f){};

    const float* wrow = w + (size_t)(i0 + lm) * KTOT + 2 * half;

    for (int kk = 0; kk < KTOT / 4; ++kk) {
        // A fragment: 16x4 f32, lanes 0-15 hold K={0,1}, lanes 16-31 K={2,3}
        v2f a2;
        a2.x = wrow[4 * kk + 0];
        a2.y = wrow[4 * kk + 1];

        // LDS row bases for the two K values this lane-half owns
        const int K0 = 4 * kk + 2 * half;
        const int j0 = K0 / 3,        t0 = K0 - 3 * j0;
        const int K1 = K0 + 1;
        const int j1 = K1 / 3,        t1 = K1 - 3 * j1;
        const int row0 = (t0 * CIN + j0) * NCOL;
        const int row1 = (t1 * CIN + j1) * NCOL;

        #pragma unroll
        for (int ct = 0; ct < NT; ++ct) {
            const int col = ct * 16 + lm;
            v2f b2;
            b2.x = sX[row0 + col];
            b2.y = sX[row1 + col];
            // (neg_a, A, neg_b, B, c_mod, C, reuse_a, reuse_b)
            acc[ct] = __builtin_amdgcn_wmma_f32_16x16x4_f32(
                false, a2, false, b2, (short)0, acc[ct], false, false);
        }
    }

    // ---- epilogue: W-roll folded into scatter, NT stores bypass L2 RT ----
    #pragma unroll
    for (int ct = 0; ct < NT; ++ct) {
        const int col = ct * 16 + lm;
        const int o   = col / WW;
        const int m   = col - o * WW;
        const int hp  = 2 * n + o;
        int wp = m + 1; if (wp >= WW) wp -= WW;   // jnp.roll(y, 1, axis=3)
        float* op = out + ((size_t)(bb * COUT + i0 + 8 * half) * HH + hp) * WW + wp;
        #pragma unroll
        for (int v = 0; v < 8; ++v) {            // D: VGPR v -> row v + 8*half
            __builtin_nontemporal_store(acc[ct][v], op + (size_t)v * HH * WW);
        }
    }
}

extern "C" void kernel_launch(void* const* d_in, const int* in_sizes, int n_in,
                              void* d_out, int out_size, void* d_ws, size_t ws_size,
                              hipStream_t stream) {
    const float* x = (const float*)d_in[0];   // (64,48,56,56) f32
    const float* w = (const float*)d_in[1];   // (192,48,3)    f32
    float* out = (float*)d_out;               // (64,192,56,56) f32

    dim3 grid(BATCH * QQ);    // 1792 blocks, one (b, n) slab each
    dim3 block(THREADS);      // 12 waves of 32
    shiftconv_wmma_f32<<<grid, block, 0, stream>>>(x, w, out);
}